// DDG_4629974745263
// MI455X (gfx1250) — compile-verified
//
#include <hip/hip_runtime.h>
#include <math.h>

// ---------------- problem constants (match reference) ----------------
#define BB 4
#define LL 2048
#define ITERS 50
#define LR 0.1f
#define MOM 0.9f
#define NC3 (BB * LL * 3)       // 24576 coord elements

typedef __attribute__((ext_vector_type(2))) float v2f;
typedef __attribute__((ext_vector_type(8))) float v8f;

// ---------------- deterministic init (JAX threefry stand-in) ----------------
__device__ __forceinline__ unsigned hashu(unsigned x) {
    x ^= x >> 16; x *= 0x7feb352du;
    x ^= x >> 15; x *= 0x846ca68bu;
    x ^= x >> 16; return x;
}

__global__ void init_kernel(float* __restrict__ c, float* __restrict__ buf,
                            float* __restrict__ g, float* __restrict__ denom) {
    int idx = blockIdx.x * blockDim.x + threadIdx.x;
    if (idx < NC3) {
        unsigned h1 = hashu(0x9E3779B9u ^ (unsigned)idx);
        unsigned h2 = hashu(0x85EBCA6Bu + (unsigned)idx * 0x27d4eb2fu);
        float u1 = (float)(h1 >> 8) * (1.0f / 16777216.0f) + 1e-7f;
        float u2 = (float)(h2 >> 8) * (1.0f / 16777216.0f);
        float r = __builtin_amdgcn_sqrtf(-2.0f * logf(u1));
        c[idx]   = r * cosf(6.28318530718f * u2);   // ~N(0,1)
        buf[idx] = 0.0f;
        g[idx]   = 0.0f;
    }
    if (idx == 0) denom[0] = 0.0f;
}

// ---------------- denom = sum(mask) ----------------
__global__ __launch_bounds__(256) void mask_sum_kernel(const float* __restrict__ mask,
                                                       float* __restrict__ denom, int n) {
    __shared__ float red[8];
    int idx = blockIdx.x * blockDim.x + threadIdx.x;
    float s = 0.0f;
    for (int i = idx; i < n; i += gridDim.x * blockDim.x) s += mask[i];
    for (int off = 16; off > 0; off >>= 1) s += __shfl_down(s, off);
    int wave = threadIdx.x >> 5;
    if ((threadIdx.x & 31) == 0) red[wave] = s;
    __syncthreads();
    if (threadIdx.x == 0) {
        float t = 0.0f;
        #pragma unroll
        for (int w = 0; w < 8; ++w) t += red[w];
        atomicAdd(denom, t);
    }
}

// ---------------- gradient kernel: one wave per unordered tile pair ----------------
// For tile pair (it <= jt):
//   WMMA#1: SQ = A(c_i) x B(-2 c_j) + C(|c_i|^2 + |c_j|^2)         (sq-dist tile)
//   s_ij   = (dL/dd_ij + dL/dd_ji) / d                              (symmetric)
//   WMMA#2: acc_row += S  x [c_j | 1]  -> g_i = rowsum*c_i - cols   (i-rows)
//   WMMA#3: acc_col += S^T x [c_i | 1]  -> g_j = rowsum*c_j - cols  (j-rows, it!=jt)
// Diagonal tiles keep only the row contribution (both pair orders already in tile).
__global__ __launch_bounds__(32) void grad_kernel(
    const float* __restrict__ conf, const float* __restrict__ pred,
    const float* __restrict__ mask, const float* __restrict__ c,
    float* __restrict__ g, const float* __restrict__ denomp)
{
    const int it = blockIdx.x;
    const int jt = blockIdx.y;
    if (jt < it) return;                      // upper-triangular tile pairs only
    const int b  = blockIdx.z;

    __shared__ float ci[16][3];
    __shared__ float sqni[16];
    __shared__ float cj[16][3];
    __shared__ float sqnj[16];
    __shared__ float s_lds[16][17];           // 17 pitch: conflict-free both row & col access
    __shared__ float gd[16][4];

    const int lane = threadIdx.x;
    const int hi   = lane >> 4;               // 0: lanes 0-15, 1: lanes 16-31
    const int n    = lane & 15;
    const int i0   = it * 16;
    const int j0   = jt * 16;

    const size_t plane = (size_t)b * LL * LL;
    const float* cb = c + (size_t)b * LL * 3;

    // prefetch the 5 tile row-sets while we set up coords (global_prefetch_b8)
    if (lane < 16) {
        __builtin_prefetch(conf + plane + (size_t)(i0 + lane) * LL + j0, 0, 1);
        __builtin_prefetch(pred + plane + (size_t)(i0 + lane) * LL + j0, 0, 1);
        __builtin_prefetch(mask + plane + (size_t)(i0 + lane) * LL + j0, 0, 1);
    } else {
        __builtin_prefetch(conf + plane + (size_t)(j0 + n) * LL + i0, 0, 1);
        __builtin_prefetch(mask + plane + (size_t)(j0 + n) * LL + i0, 0, 1);
    }

    // lanes 0-15 stage c_i rows, lanes 16-31 stage c_j rows (concurrent)
    if (lane < 16) {
        float x = cb[(i0 + lane) * 3 + 0];
        float y = cb[(i0 + lane) * 3 + 1];
        float z = cb[(i0 + lane) * 3 + 2];
        ci[lane][0] = x; ci[lane][1] = y; ci[lane][2] = z;
        sqni[lane] = x * x + y * y + z * z;
    } else {
        float x = cb[(j0 + n) * 3 + 0];
        float y = cb[(j0 + n) * 3 + 1];
        float z = cb[(j0 + n) * 3 + 2];
        cj[n][0] = x; cj[n][1] = y; cj[n][2] = z;
        sqnj[n] = x * x + y * y + z * z;
    }
    __syncthreads();

    const float inv_denom = 1.0f / (denomp[0] + 1e-6f);

    // WMMA#1 operands (A 16x4: lane row = n; lanes<16 K=0,1; lanes>=16 K=2,3)
    v2f A1, B1;
    A1.x = hi ? ci[n][2] : ci[n][0];
    A1.y = hi ? 0.0f     : ci[n][1];
    B1.x = hi ? (-2.0f * cj[n][2]) : (-2.0f * cj[n][0]);
    B1.y = hi ? 0.0f               : (-2.0f * cj[n][1]);

    v8f C1;
    #pragma unroll
    for (int v = 0; v < 8; ++v) C1[v] = sqni[v + 8 * hi] + sqnj[n];

    v8f SQ = __builtin_amdgcn_wmma_f32_16x16x4_f32(
        false, A1, false, B1, (short)0, C1, false, false);

    // per-element symmetric pair weight s_ij
    #pragma unroll
    for (int v = 0; v < 8; ++v) {
        const int m   = v + 8 * hi;
        const int row = i0 + m;
        const int col = j0 + n;
        float sq = SQ[v];
        sq = sq > 0.0f ? sq : 0.0f;
        const float d = __builtin_amdgcn_sqrtf(sq);   // v_sqrt_f32

        const size_t oij = plane + (size_t)row * LL + col;
        const size_t oji = plane + (size_t)col * LL + row;
        const float cf_ij = conf[oij];
        const float pr    = pred[oij];                // pred symmetric by construction
        const float mk_ij = mask[oij];
        const float cf_ji = conf[oji];
        const float mk_ji = mask[oji];

        float w = 2.0f * (cf_ij * mk_ij + cf_ji * mk_ji) * (d - pr) * inv_denom;
        if (row != col && d < 1.0f)
            w -= 0.1f * ((mk_ij > 0.0f ? 1.0f : 0.0f) + (mk_ji > 0.0f ? 1.0f : 0.0f));
        s_lds[m][n] = (d > 0.0f) ? (w * __builtin_amdgcn_rcpf(d)) : 0.0f;  // v_rcp_f32
    }
    __syncthreads();

    const bool offdiag = (it != jt);

    // WMMA#2: acc_row += S x [c_j | 1]   (4 chained K=4 chunks)
    v8f acc_row = {};
    #pragma unroll
    for (int q = 0; q < 4; ++q) {
        const int k0 = 4 * q + 2 * hi;
        v2f A2;                                        // A row = n (i-local), K = k0,k0+1 (j-local)
        A2.x = s_lds[n][k0 + 0];
        A2.y = s_lds[n][k0 + 1];
        v2f B2;                                        // B row jj = k0,k0+1, col = n
        B2.x = (n < 3) ? cj[k0][n]     : (n == 3 ? 1.0f : 0.0f);
        B2.y = (n < 3) ? cj[k0 + 1][n] : (n == 3 ? 1.0f : 0.0f);
        acc_row = __builtin_amdgcn_wmma_f32_16x16x4_f32(
            false, A2, false, B2, (short)0, acc_row, false, false);
    }

    // WMMA#3: acc_col += S^T x [c_i | 1]  (off-diagonal tiles only; block-uniform branch)
    v8f acc_col = {};
    if (offdiag) {
        #pragma unroll
        for (int q = 0; q < 4; ++q) {
            const int k0 = 4 * q + 2 * hi;
            v2f A3;                                    // S^T: row = n (j-local), K = k0 (i-local)
            A3.x = s_lds[k0 + 0][n];
            A3.y = s_lds[k0 + 1][n];
            v2f B3;                                    // B row ii = k0,k0+1, col = n
            B3.x = (n < 3) ? ci[k0][n]     : (n == 3 ? 1.0f : 0.0f);
            B3.y = (n < 3) ? ci[k0 + 1][n] : (n == 3 ? 1.0f : 0.0f);
            acc_col = __builtin_amdgcn_wmma_f32_16x16x4_f32(
                false, A3, false, B3, (short)0, acc_col, false, false);
        }
    }

    // scatter row contribution: g_i += rowsum*c_i - (sum_j s*c_j)
    __syncthreads();
    if (n < 4) {
        #pragma unroll
        for (int v = 0; v < 8; ++v) gd[v + 8 * hi][n] = acc_row[v];
    }
    __syncthreads();
    if (lane < 16) {
        const float rs = gd[lane][3];
        float* gp = g + ((size_t)b * LL + i0 + lane) * 3;
        atomicAdd(&gp[0], rs * ci[lane][0] - gd[lane][0]);
        atomicAdd(&gp[1], rs * ci[lane][1] - gd[lane][1]);
        atomicAdd(&gp[2], rs * ci[lane][2] - gd[lane][2]);
    }

    // scatter column contribution: g_j += colsum*c_j - (sum_i s*c_i)
    if (offdiag) {
        __syncthreads();
        if (n < 4) {
            #pragma unroll
            for (int v = 0; v < 8; ++v) gd[v + 8 * hi][n] = acc_col[v];
        }
        __syncthreads();
        if (lane < 16) {
            const float rs = gd[lane][3];
            float* gp = g + ((size_t)b * LL + j0 + lane) * 3;
            atomicAdd(&gp[0], rs * cj[lane][0] - gd[lane][0]);
            atomicAdd(&gp[1], rs * cj[lane][1] - gd[lane][1]);
            atomicAdd(&gp[2], rs * cj[lane][2] - gd[lane][2]);
        }
    }
}

// ---------------- SGD momentum update; consumes and re-zeroes g ----------------
__global__ void update_kernel(float* __restrict__ c, float* __restrict__ buf,
                              float* __restrict__ g, float* __restrict__ out) {
    int idx = blockIdx.x * blockDim.x + threadIdx.x;
    if (idx < NC3) {
        float bv = MOM * buf[idx] + g[idx];
        buf[idx] = bv;
        float nc = c[idx] - LR * bv;
        c[idx] = nc;
        out[idx] = nc;
        g[idx] = 0.0f;      // ready for next iteration's atomic accumulation
    }
}

extern "C" void kernel_launch(void* const* d_in, const int* in_sizes, int n_in,
                              void* d_out, int out_size, void* d_ws, size_t ws_size,
                              hipStream_t stream) {
    const float* pred = (const float*)d_in[0];   // pred_dist [B,L,L]
    const float* conf = (const float*)d_in[1];   // confidence [B,L,L]
    const float* mask = (const float*)d_in[2];   // mask [B,L,L]

    float* ws  = (float*)d_ws;
    float* c   = ws;                 // [B*L*3]
    float* buf = ws + NC3;           // [B*L*3]
    float* g   = ws + 2 * NC3;       // [B*L*3]
    float* den = ws + 3 * NC3;       // [1]
    float* out = (float*)d_out;

    init_kernel<<<(NC3 + 255) / 256, 256, 0, stream>>>(c, buf, g, den);
    mask_sum_kernel<<<512, 256, 0, stream>>>(mask, den, BB * LL * LL);

    dim3 grid(LL / 16, LL / 16, BB);             // upper triangle active
    for (int it = 0; it < ITERS; ++it) {
        grad_kernel<<<grid, 32, 0, stream>>>(conf, pred, mask, c, g, den);
        update_kernel<<<(NC3 + 255) / 256, 256, 0, stream>>>(c, buf, g, out);
    }
}